// NetworkMamba_11854109737406
// MI455X (gfx1250) — compile-verified
//
#include <hip/hip_runtime.h>

#define D_MODEL 32
#define D_STATE 16
#define D_CONV 4
#define D_INNER 64
#define DT_RANK 2
#define BATCH 64
#define C_IN 64
#define L_SEQ 1024
#define FEAT 128
#define XDBL_STRIDE 48

typedef __attribute__((ext_vector_type(2))) float v2f;
typedef __attribute__((ext_vector_type(8))) float v8f;
typedef __attribute__((ext_vector_type(4))) int v4i;
typedef __attribute__((ext_vector_type(8))) int v8i;

// D(16x16,f32) = A(16x4,f32) x B(4x16,f32) + C ; full fp32 matrix core
static __device__ __forceinline__ v8f wmma_f32x4(v2f a, v2f b, v8f c) {
  return __builtin_amdgcn_wmma_f32_16x16x4_f32(false, a, false, b, (short)0, c,
                                               false, false);
}

static __device__ __forceinline__ float silu_f(float x) {
  return x / (1.0f + expf(-x));
}

// ---------------------------------------------------------------------------
// Kernel 1: h = conv_w(32x64) @ x[b](64x1024) ; BN ; exact GELU ; -> u (bl,32)
// One wave computes a 32(d) x 16(l) tile.
// ---------------------------------------------------------------------------
__global__ __launch_bounds__(32) void k_conv_bn_gelu(
    const float* __restrict__ X, const float* __restrict__ CW,
    const float* __restrict__ CB, const float* __restrict__ G,
    const float* __restrict__ Bb, const float* __restrict__ Mn,
    const float* __restrict__ Vr, float* __restrict__ U) {
  const int b = blockIdx.y;
  const int l0 = blockIdx.x * 16;
  const int lane = threadIdx.x;
  const int g = lane >> 4, m = lane & 15;
  const float* xb = X + b * (C_IN * L_SEQ);
  v8f acc0 = {}; v8f acc1 = {};
#pragma unroll
  for (int kk = 0; kk < 16; ++kk) {
    const int c0 = kk * 4 + 2 * g;            // this lane-group's K pair
    v2f bf; bf.x = xb[c0 * L_SEQ + l0 + m];
            bf.y = xb[(c0 + 1) * L_SEQ + l0 + m];
    v2f a0; a0.x = CW[m * C_IN + c0];         a0.y = CW[m * C_IN + c0 + 1];
    v2f a1; a1.x = CW[(16 + m) * C_IN + c0];  a1.y = CW[(16 + m) * C_IN + c0 + 1];
    acc0 = wmma_f32x4(a0, bf, acc0);
    acc1 = wmma_f32x4(a1, bf, acc1);
  }
  const int l = l0 + m;
#pragma unroll
  for (int r = 0; r < 8; ++r) {
    const int dA = r + 8 * g;
    const int dB = 16 + r + 8 * g;
    {
      float sc = G[dA] * rsqrtf(Vr[dA] + 1e-5f);
      float h = (acc0[r] + CB[dA] - Mn[dA]) * sc + Bb[dA];
      h = 0.5f * h * (1.0f + erff(h * 0.70710678118654752f));
      U[(b * L_SEQ + l) * D_MODEL + dA] = h;
    }
    {
      float sc = G[dB] * rsqrtf(Vr[dB] + 1e-5f);
      float h = (acc1[r] + CB[dB] - Mn[dB]) * sc + Bb[dB];
      h = 0.5f * h * (1.0f + erff(h * 0.70710678118654752f));
      U[(b * L_SEQ + l) * D_MODEL + dB] = h;
    }
  }
}

// ---------------------------------------------------------------------------
// Kernel 2: xz = u(65536x32) @ in_proj_w^T(32x128).
// in_proj_w (128x32 f32, 16KB contiguous) is staged into LDS with one
// Tensor-Data-Mover descriptor per workgroup, then WMMA B-fragments are fed
// from LDS. xm half stored transposed (b,d,L) for the depthwise conv; z half
// stored (bl,d).
// ---------------------------------------------------------------------------
__global__ __launch_bounds__(32) void k_in_proj(
    const float* __restrict__ U, const float* __restrict__ W,
    float* __restrict__ XMT, float* __restrict__ Z) {
  __shared__ float sW[FEAT * D_MODEL];  // 16 KB

  // ---- Tensor DMA descriptor (D#), CDNA5 ISA ----
  const unsigned long long ga = (unsigned long long)(uintptr_t)W;
  const unsigned int lds_addr = (unsigned int)(uintptr_t)sW;  // addr[31:0] = LDS offset
  v4i g0;
  g0[0] = 1;                                   // count=1, user descriptor
  g0[1] = (int)lds_addr;                       // [63:32]  lds_addr
  g0[2] = (int)(unsigned int)ga;               // [95:64]  global_addr[31:0]
  g0[3] = (int)((((unsigned int)(ga >> 32)) & 0x01FFFFFFu) |
                0x80000000u);                  // [120:96] addr[56:32], [127:126] type=2
  v8i g1;
  g1[0] = (int)(2u << 16);    // data_size=2 (4 bytes); no multicast/pad/iterate
  g1[1] = (int)(32u << 16);   // [63:48]  tensor_dim0 = 32
  g1[2] = (int)(128u << 16);  // [95:80]  tensor_dim1 = 128
  g1[3] = (int)(32u << 16);   // [127:112] tile_dim0 = 32
  g1[4] = 128;                // [143:128] tile_dim1 = 128
  g1[5] = 32;                 // [207:160] tensor_dim0_stride = 32
  g1[6] = 0;
  g1[7] = 0;
  asm volatile("tensor_load_to_lds %0, %1" :: "s"(g0), "s"(g1) : "memory");
  __builtin_amdgcn_s_wait_tensorcnt(0);

  const int row0 = blockIdx.x * 16;
  const int lane = threadIdx.x;
  const int g = lane >> 4, m = lane & 15;
  v8f acc[8];
#pragma unroll
  for (int i = 0; i < 8; ++i) acc[i] = (v8f){};
#pragma unroll
  for (int kk = 0; kk < 8; ++kk) {
    const int k0 = kk * 4 + 2 * g;
    v2f a; a.x = U[(row0 + m) * D_MODEL + k0];
           a.y = U[(row0 + m) * D_MODEL + k0 + 1];
#pragma unroll
    for (int nt = 0; nt < 8; ++nt) {
      const int n = nt * 16 + m;
      v2f bf; bf.x = sW[n * D_MODEL + k0]; bf.y = sW[n * D_MODEL + k0 + 1];
      acc[nt] = wmma_f32x4(a, bf, acc[nt]);
    }
  }
#pragma unroll
  for (int nt = 0; nt < 8; ++nt) {
    const int n = nt * 16 + m;
#pragma unroll
    for (int r = 0; r < 8; ++r) {
      const int bl = row0 + r + 8 * g;
      const float v = acc[nt][r];
      if (nt < 4) {
        const int bb = bl >> 10, ll = bl & (L_SEQ - 1);
        XMT[(bb * D_INNER + n) * L_SEQ + ll] = v;   // pre-conv xm, (b,d,L)
      } else {
        Z[bl * D_INNER + (n - D_INNER)] = v;        // gate z, (bl,d)
      }
    }
  }
}

// ---------------------------------------------------------------------------
// Kernel 3a: causal depthwise conv1d (k=4) + bias + SiLU -> xm (bl,d)
// ---------------------------------------------------------------------------
__global__ void k_conv1d_silu(const float* __restrict__ XMT,
                              const float* __restrict__ W1,
                              const float* __restrict__ B1,
                              float* __restrict__ XM) {
  const int idx = blockIdx.x * blockDim.x + threadIdx.x;
  if (idx >= BATCH * D_INNER * L_SEQ) return;
  const int l = idx & (L_SEQ - 1);
  const int d = (idx >> 10) & (D_INNER - 1);
  const int b = idx >> 16;
  const float* src = XMT + (b * D_INNER + d) * L_SEQ;
  float s = B1[d];
#pragma unroll
  for (int t = 0; t < D_CONV; ++t) {
    const int ls = l + t - (D_CONV - 1);
    if (ls >= 0) s += W1[d * D_CONV + t] * src[ls];
  }
  XM[(b * L_SEQ + l) * D_INNER + d] = silu_f(s);
}

// ---------------------------------------------------------------------------
// Kernel 3b: x_dbl = xm(65536x64) @ x_proj_w^T(64x34), N padded to 48.
// ---------------------------------------------------------------------------
__global__ __launch_bounds__(32) void k_x_proj(
    const float* __restrict__ XM, const float* __restrict__ XP,
    float* __restrict__ XDBL) {
  const int row0 = blockIdx.x * 16;
  const int lane = threadIdx.x;
  const int g = lane >> 4, m = lane & 15;
  v8f acc[3]; acc[0] = (v8f){}; acc[1] = (v8f){}; acc[2] = (v8f){};
#pragma unroll
  for (int kk = 0; kk < 16; ++kk) {
    const int k0 = kk * 4 + 2 * g;
    v2f a; a.x = XM[(row0 + m) * D_INNER + k0];
           a.y = XM[(row0 + m) * D_INNER + k0 + 1];
#pragma unroll
    for (int nt = 0; nt < 3; ++nt) {
      const int n = nt * 16 + m;
      v2f bf;
      bf.x = (n < 34) ? XP[n * D_INNER + k0] : 0.0f;
      bf.y = (n < 34) ? XP[n * D_INNER + k0 + 1] : 0.0f;
      acc[nt] = wmma_f32x4(a, bf, acc[nt]);
    }
  }
#pragma unroll
  for (int nt = 0; nt < 3; ++nt) {
    const int n = nt * 16 + m;
    if (n < 34) {
#pragma unroll
      for (int r = 0; r < 8; ++r) {
        const int bl = row0 + r + 8 * g;
        XDBL[bl * XDBL_STRIDE + n] = acc[nt][r];
      }
    }
  }
}

// ---------------------------------------------------------------------------
// Kernel 3c: dt = softplus(x_dbl[:, :2] @ dt_proj_w^T + dt_proj_b) (rank-2)
// ---------------------------------------------------------------------------
__global__ void k_dt(const float* __restrict__ XDBL,
                     const float* __restrict__ DTW,
                     const float* __restrict__ DTB, float* __restrict__ DT) {
  const int idx = blockIdx.x * blockDim.x + threadIdx.x;
  if (idx >= BATCH * L_SEQ * D_INNER) return;
  const int d = idx & (D_INNER - 1);
  const int bl = idx >> 6;
  const float x0 = XDBL[bl * XDBL_STRIDE + 0];
  const float x1 = XDBL[bl * XDBL_STRIDE + 1];
  const float v = x0 * DTW[d * 2] + x1 * DTW[d * 2 + 1] + DTB[d];
  DT[idx] = (v > 20.0f) ? v : log1pf(expf(v));
}

// ---------------------------------------------------------------------------
// Kernel 4: selective scan. One wave owns (batch b, 2 d-channels x 16 states).
// The only true serial dependence is h = g*h + c (one FMA/step); exp, loads,
// and the 16-lane shuffle reduction are off the chain. Prefetch streams 8
// steps ahead (global_prefetch_b8) and unroll x2 to let the compiler pipeline.
// Fused: y = (scan + u*D) * silu(z).
// ---------------------------------------------------------------------------
__global__ __launch_bounds__(32) void k_scan(
    const float* __restrict__ XM, const float* __restrict__ DT,
    const float* __restrict__ XDBL, const float* __restrict__ Z,
    const float* __restrict__ Alog, const float* __restrict__ Dp,
    float* __restrict__ Y) {
  const int b = blockIdx.y;
  const int lane = threadIdx.x;
  const int g = lane >> 4, n = lane & 15;
  const int d = blockIdx.x * 2 + g;
  const float Adn = -expf(Alog[d * D_STATE + n]);
  const float Dd = Dp[d];
  float h = 0.0f;
#pragma unroll 2
  for (int l = 0; l < L_SEQ; ++l) {
    const int bl = b * L_SEQ + l;
    if (l + 8 < L_SEQ) {
      const int blp = bl + 8;
      __builtin_prefetch(&XDBL[blp * XDBL_STRIDE + 2 + n], 0, 3);
      __builtin_prefetch(&XM[blp * D_INNER + d], 0, 3);
      __builtin_prefetch(&DT[blp * D_INNER + d], 0, 3);
    }
    const float u = XM[bl * D_INNER + d];
    const float dtv = DT[bl * D_INNER + d];
    const float Bv = XDBL[bl * XDBL_STRIDE + 2 + n];
    const float Cv = XDBL[bl * XDBL_STRIDE + 18 + n];
    h = expf(dtv * Adn) * h + (dtv * u) * Bv;
    float p = h * Cv;
    p += __shfl_xor(p, 8, 32);
    p += __shfl_xor(p, 4, 32);
    p += __shfl_xor(p, 2, 32);
    p += __shfl_xor(p, 1, 32);
    if (n == 0) {
      const float z = Z[bl * D_INNER + d];
      Y[bl * D_INNER + d] = (p + u * Dd) * silu_f(z);
    }
  }
}

// ---------------------------------------------------------------------------
// Kernel 5: out = y(65536x64) @ out_proj_w^T(64x32), fused mean over L
// (partial column sums per tile -> atomicAdd into pooled, pre-scaled 1/1024).
// ---------------------------------------------------------------------------
__global__ __launch_bounds__(32) void k_out_proj_pool(
    const float* __restrict__ Y, const float* __restrict__ OP,
    float* __restrict__ POOL) {
  const int row0 = blockIdx.x * 16;
  const int b = row0 >> 10;  // 16-row tiles never cross a batch (1024 % 16 == 0)
  const int lane = threadIdx.x;
  const int g = lane >> 4, m = lane & 15;
  v8f acc[2]; acc[0] = (v8f){}; acc[1] = (v8f){};
#pragma unroll
  for (int kk = 0; kk < 16; ++kk) {
    const int k0 = kk * 4 + 2 * g;
    v2f a; a.x = Y[(row0 + m) * D_INNER + k0];
           a.y = Y[(row0 + m) * D_INNER + k0 + 1];
#pragma unroll
    for (int nt = 0; nt < 2; ++nt) {
      const int n = nt * 16 + m;
      v2f bf; bf.x = OP[n * D_INNER + k0]; bf.y = OP[n * D_INNER + k0 + 1];
      acc[nt] = wmma_f32x4(a, bf, acc[nt]);
    }
  }
#pragma unroll
  for (int nt = 0; nt < 2; ++nt) {
    float s = 0.0f;
#pragma unroll
    for (int r = 0; r < 8; ++r) s += acc[nt][r];   // rows r+8g of this tile
    s += __shfl_xor(s, 16, 32);                    // combine both row halves
    if (g == 0)
      atomicAdd(&POOL[b * D_MODEL + nt * 16 + m], s * (1.0f / (float)L_SEQ));
  }
}

__global__ void k_zero(float* __restrict__ p, int n) {
  const int i = blockIdx.x * blockDim.x + threadIdx.x;
  if (i < n) p[i] = 0.0f;
}

// ---------------------------------------------------------------------------
// Kernel 6: final fc: (64x32) @ (32x128)^T + b  -> d_out (64x128)
// ---------------------------------------------------------------------------
__global__ void k_fc(const float* __restrict__ POOL, const float* __restrict__ FW,
                     const float* __restrict__ FB, float* __restrict__ OUT) {
  const int i = blockIdx.x * blockDim.x + threadIdx.x;
  if (i >= BATCH * FEAT) return;
  const int f = i & (FEAT - 1);
  const int b = i >> 7;
  float s = FB[f];
#pragma unroll
  for (int k = 0; k < D_MODEL; ++k)
    s += POOL[b * D_MODEL + k] * FW[f * D_MODEL + k];
  OUT[i] = s;
}

extern "C" void kernel_launch(void* const* d_in, const int* in_sizes, int n_in,
                              void* d_out, int out_size, void* d_ws, size_t ws_size,
                              hipStream_t stream) {
  (void)in_sizes; (void)n_in; (void)out_size; (void)ws_size;
  const float* X   = (const float*)d_in[0];   // (64,64,32,32)
  const float* CW  = (const float*)d_in[1];   // (32,64)
  const float* CB  = (const float*)d_in[2];   // (32)
  const float* G   = (const float*)d_in[3];   // bn_gamma
  const float* Bb  = (const float*)d_in[4];   // bn_beta
  const float* Mn  = (const float*)d_in[5];   // bn_mean
  const float* Vr  = (const float*)d_in[6];   // bn_var
  const float* IPW = (const float*)d_in[7];   // (128,32)
  const float* C1W = (const float*)d_in[8];   // (64,4)
  const float* C1B = (const float*)d_in[9];   // (64)
  const float* XPW = (const float*)d_in[10];  // (34,64)
  const float* DTW = (const float*)d_in[11];  // (64,2)
  const float* DTB = (const float*)d_in[12];  // (64)
  const float* AL  = (const float*)d_in[13];  // (64,16)
  const float* DP  = (const float*)d_in[14];  // (64)
  const float* OPW = (const float*)d_in[15];  // (32,64)
  const float* FW  = (const float*)d_in[16];  // (128,32)
  const float* FB  = (const float*)d_in[17];  // (128)
  float* OUT = (float*)d_out;

  float* ws   = (float*)d_ws;
  float* U    = ws;                 // 65536*32
  float* XMT  = U    + 2097152;     // 64*64*1024
  float* Z    = XMT  + 4194304;     // 65536*64
  float* XM   = Z    + 4194304;     // 65536*64
  float* XDBL = XM   + 4194304;     // 65536*48
  float* DT   = XDBL + 3145728;     // 65536*64
  float* Y    = DT   + 4194304;     // 65536*64
  float* POOL = Y    + 4194304;     // 64*32

  k_conv_bn_gelu<<<dim3(L_SEQ / 16, BATCH), 32, 0, stream>>>(X, CW, CB, G, Bb,
                                                             Mn, Vr, U);
  k_in_proj<<<(BATCH * L_SEQ) / 16, 32, 0, stream>>>(U, IPW, XMT, Z);
  k_conv1d_silu<<<(BATCH * D_INNER * L_SEQ) / 256, 256, 0, stream>>>(XMT, C1W,
                                                                     C1B, XM);
  k_x_proj<<<(BATCH * L_SEQ) / 16, 32, 0, stream>>>(XM, XPW, XDBL);
  k_dt<<<(BATCH * L_SEQ * D_INNER) / 256, 256, 0, stream>>>(XDBL, DTW, DTB, DT);
  k_scan<<<dim3(D_INNER / 2, BATCH), 32, 0, stream>>>(XM, DT, XDBL, Z, AL, DP, Y);
  k_zero<<<(BATCH * D_MODEL + 255) / 256, 256, 0, stream>>>(POOL,
                                                            BATCH * D_MODEL);
  k_out_proj_pool<<<(BATCH * L_SEQ) / 16, 32, 0, stream>>>(Y, OPW, POOL);
  k_fc<<<(BATCH * FEAT + 255) / 256, 256, 0, stream>>>(POOL, FW, FB, OUT);
}